// UncertaintyAwareFNO_49804440764636
// MI455X (gfx1250) — compile-verified
//
#include <hip/hip_runtime.h>
#include <math.h>

// ---- problem constants (match reference) ----
#define B_      8
#define S_      32
#define CIN_    4
#define W_      64
#define NL_     2
#define NM_     3
#define MODE_   16
#define G_      8
#define S3_     (S_*S_*S_)              // 32768
#define NMODES_ (MODE_*MODE_*MODE_)     // 4096
#define TWOPI_OVER_32 0.19634954084936207f

typedef float v2f __attribute__((ext_vector_type(2)));
typedef float v8f __attribute__((ext_vector_type(8)));

__device__ __forceinline__ float gelu_tanh(float x) {
    float x3 = x * x * x;
    return 0.5f * x * (1.0f + tanhf(0.7978845608028654f * (x + 0.044715f * x3)));
}

// ---------------------------------------------------------------------------
// K0: embedding  h[b][w][xyz] = sum_i x[b,xyz,i]*emb_w[w,i] + emb_b[w]
// grid (S3/256, B), block 256. x read once (float4/pos), h written coalesced.
// ---------------------------------------------------------------------------
__global__ void k_embed(const float* __restrict__ x, const float* __restrict__ emb_w,
                        const float* __restrict__ emb_b, float* __restrict__ h) {
    __shared__ float ew[W_ * CIN_];
    __shared__ float eb[W_];
    int tid = threadIdx.x;
    ew[tid] = emb_w[tid];                    // 256 == 64*4
    if (tid < W_) eb[tid] = emb_b[tid];
    __syncthreads();
    int b  = blockIdx.y;
    int sp = blockIdx.x * 256 + tid;
    const float4 xv = *(const float4*)(x + ((size_t)b * S3_ + sp) * CIN_);
    float* hb = h + (size_t)b * W_ * S3_ + sp;
    #pragma unroll 4
    for (int w = 0; w < W_; ++w) {
        float v = eb[w];
        v = fmaf(xv.x, ew[w*4+0], v);
        v = fmaf(xv.y, ew[w*4+1], v);
        v = fmaf(xv.z, ew[w*4+2], v);
        v = fmaf(xv.w, ew[w*4+3], v);
        hb[(size_t)w * S3_] = v;
    }
}

// ---------------------------------------------------------------------------
// K1: per-(b,c) sum and sum-of-squares over 32768 spatial elems
// grid 512 blocks x 256 threads
// ---------------------------------------------------------------------------
__global__ void k_stats(const float* __restrict__ h, float* __restrict__ chanSum,
                        float* __restrict__ chanSqs) {
    __shared__ float ls[256], ls2[256];
    int bc = blockIdx.x;
    const float* p = h + (size_t)bc * S3_;
    float s = 0.f, s2 = 0.f;
    for (int i = threadIdx.x; i < S3_; i += 256) {
        float v = p[i];
        s += v;
        s2 = fmaf(v, v, s2);
    }
    ls[threadIdx.x] = s; ls2[threadIdx.x] = s2;
    __syncthreads();
    for (int st = 128; st > 0; st >>= 1) {
        if (threadIdx.x < st) {
            ls[threadIdx.x]  += ls[threadIdx.x + st];
            ls2[threadIdx.x] += ls2[threadIdx.x + st];
        }
        __syncthreads();
    }
    if (threadIdx.x == 0) { chanSum[bc] = ls[0]; chanSqs[bc] = ls2[0]; }
}

// ---------------------------------------------------------------------------
// K2: group-norm params + attention mixture weights (tiny, one block)
// ---------------------------------------------------------------------------
__global__ void k_gn_attn(const float* __restrict__ chanSum, const float* __restrict__ chanSqs,
                          const float* __restrict__ gamma, const float* __restrict__ beta,
                          const float* __restrict__ w1, const float* __restrict__ b1,
                          const float* __restrict__ w2, const float* __restrict__ b2,
                          float* __restrict__ gnScale, float* __restrict__ gnShift,
                          float* __restrict__ mw) {
    __shared__ float pooled[B_ * W_];
    __shared__ float act[B_ * W_];
    __shared__ float mu_s[B_ * G_], rsig_s[B_ * G_];
    __shared__ float logit[B_ * NM_];
    int tid = threadIdx.x;
    for (int idx = tid; idx < B_ * W_; idx += 256)
        pooled[idx] = chanSum[idx] * (1.0f / (float)S3_);
    __syncthreads();
    if (tid < B_ * G_) {
        int b = tid >> 3, g = tid & 7;
        float s = 0.f, s2 = 0.f;
        for (int c = g * 8; c < g * 8 + 8; ++c) {
            s  += chanSum[b * W_ + c];
            s2 += chanSqs[b * W_ + c];
        }
        float inv = 1.0f / (8.0f * (float)S3_);
        float mu  = s * inv;
        float var = s2 * inv - mu * mu;
        mu_s[tid]   = mu;
        rsig_s[tid] = rsqrtf(var + 1e-5f);
    }
    __syncthreads();
    for (int idx = tid; idx < B_ * W_; idx += 256) {
        int b = idx >> 6, c = idx & 63, g = c >> 3;
        float sc = gamma[c] * rsig_s[b * G_ + g];
        gnScale[idx] = sc;
        gnShift[idx] = beta[c] - mu_s[b * G_ + g] * sc;
    }
    // attention MLP
    for (int idx = tid; idx < B_ * W_; idx += 256) {
        int b = idx >> 6, o = idx & 63;
        float s = b1[o];
        for (int i = 0; i < W_; ++i) s = fmaf(pooled[b * W_ + i], w1[o * W_ + i], s);
        act[idx] = fmaxf(s, 0.f);
    }
    __syncthreads();
    if (tid < B_ * NM_) {
        int b = tid / NM_, m = tid % NM_;
        float s = b2[m];
        for (int i = 0; i < W_; ++i) s = fmaf(act[b * W_ + i], w2[m * W_ + i], s);
        logit[tid] = s;
    }
    __syncthreads();
    if (tid < B_) {
        float l0 = logit[tid*3], l1 = logit[tid*3+1], l2 = logit[tid*3+2];
        float mx = fmaxf(l0, fmaxf(l1, l2));
        float e0 = __expf(l0 - mx), e1 = __expf(l1 - mx), e2 = __expf(l2 - mx);
        float inv = 1.0f / (e0 + e1 + e2);
        mw[tid*3] = e0 * inv; mw[tid*3+1] = e1 * inv; mw[tid*3+2] = e2 * inv;
    }
}

// ---------------------------------------------------------------------------
// K3: fused group-norm + partial DFT along z (real -> 16 complex freqs)
// out C1[bc][x][y][kz], 8.4M threads
// ---------------------------------------------------------------------------
__global__ void k_dftz(const float* __restrict__ h, const float* __restrict__ gnScale,
                       const float* __restrict__ gnShift, float2* __restrict__ out) {
    __shared__ float2 tw[32];
    if (threadIdx.x < 32) {
        float a = -TWOPI_OVER_32 * (float)threadIdx.x;
        tw[threadIdx.x] = make_float2(cosf(a), sinf(a));
    }
    __syncthreads();
    size_t id = (size_t)blockIdx.x * 256 + threadIdx.x;
    int kz = (int)(id & 15);
    size_t r = id >> 4;
    int y = (int)(r & 31); r >>= 5;
    int x = (int)(r & 31); r >>= 5;
    int bc = (int)r;
    float sc = gnScale[bc], sh = gnShift[bc];
    const float* hp = h + (size_t)bc * S3_ + x * (S_ * S_) + y * S_;
    float ar = 0.f, ai = 0.f;
    #pragma unroll 8
    for (int z = 0; z < S_; ++z) {
        float v = fmaf(hp[z], sc, sh);
        float2 w = tw[(kz * z) & 31];
        ar = fmaf(v, w.x, ar);
        ai = fmaf(v, w.y, ai);
    }
    out[id] = make_float2(ar, ai);
}

// ---------------------------------------------------------------------------
// K4: partial DFT along y (32 -> 16), complex. out C2[bc][x][ky][kz]
// ---------------------------------------------------------------------------
__global__ void k_dfty(const float2* __restrict__ in, float2* __restrict__ out) {
    __shared__ float2 tw[32];
    if (threadIdx.x < 32) {
        float a = -TWOPI_OVER_32 * (float)threadIdx.x;
        tw[threadIdx.x] = make_float2(cosf(a), sinf(a));
    }
    __syncthreads();
    size_t id = (size_t)blockIdx.x * 256 + threadIdx.x;
    int kz = (int)(id & 15);
    int ky = (int)((id >> 4) & 15);
    int x  = (int)((id >> 8) & 31);
    int bc = (int)(id >> 13);
    const float2* base = in + (((size_t)bc * S_ + x) * S_) * MODE_ + kz;
    float ar = 0.f, ai = 0.f;
    #pragma unroll 8
    for (int y = 0; y < S_; ++y) {
        float2 u = base[y * MODE_];
        float2 w = tw[(ky * y) & 31];
        ar += u.x * w.x - u.y * w.y;
        ai += u.x * w.y + u.y * w.x;
    }
    out[id] = make_float2(ar, ai);
}

// ---------------------------------------------------------------------------
// K5: partial DFT along x (32 -> 16); write mode-major Xm[p][c][b]
// ---------------------------------------------------------------------------
__global__ void k_dftx(const float2* __restrict__ in, float2* __restrict__ Xm) {
    __shared__ float2 tw[32];
    if (threadIdx.x < 32) {
        float a = -TWOPI_OVER_32 * (float)threadIdx.x;
        tw[threadIdx.x] = make_float2(cosf(a), sinf(a));
    }
    __syncthreads();
    size_t id = (size_t)blockIdx.x * 256 + threadIdx.x;
    int kz = (int)(id & 15);
    int ky = (int)((id >> 4) & 15);
    int kx = (int)((id >> 8) & 15);
    int bc = (int)(id >> 12);
    const float2* base = in + ((size_t)bc * S_ * MODE_ + ky) * MODE_ + kz;
    float ar = 0.f, ai = 0.f;
    #pragma unroll 8
    for (int x = 0; x < S_; ++x) {
        float2 u = base[(size_t)x * MODE_ * MODE_];
        float2 w = tw[(kx * x) & 31];
        ar += u.x * w.x - u.y * w.y;
        ai += u.x * w.y + u.y * w.x;
    }
    int p = (kx * MODE_ + ky) * MODE_ + kz;
    int b = bc >> 6, c = bc & 63;
    Xm[((size_t)p * W_ + c) * B_ + b] = make_float2(ar, ai);
}

// ---------------------------------------------------------------------------
// K6: spectral contraction with WMMA (V_WMMA_F32_16X16X4_F32).
// One wave per mode p; 32 consecutive modes per 1024-thread block so each
// 128B weight cacheline (32 modes) is consumed by exactly one WGP.
// Per mode:  Y[o,b] = sum_m mw[b,m] * sum_i W_m[i,o] * X[i,b]   (complex)
// A = W^T tile (M=o 16, K=i 4), B = X (K=i 4, N=b 16, b>=8 zero),
// mixture weight folded into the per-lane B fragment scale.
// grid (NMODES/32, 4 o-tiles), block 1024.
// ---------------------------------------------------------------------------
__global__ void __launch_bounds__(1024)
k_contract(const float2* __restrict__ Xm, const float* __restrict__ wr,
           const float* __restrict__ wi, const float* __restrict__ mw,
           float2* __restrict__ Ym) {
    const int wave  = threadIdx.x >> 5;
    const int lane  = threadIdx.x & 31;
    const int p     = blockIdx.x * 32 + wave;   // mode index
    const int otile = blockIdx.y;               // 0..3
    const int n     = lane & 15;                // WMMA N column (= batch b), also A row M (= o)
    const int kh    = lane >> 4;                // K half select
    const int nb    = (n < 8) ? n : 0;
    const float msk = (n < 8) ? 1.f : 0.f;
    const float s0 = mw[nb * NM_ + 0] * msk;
    const float s1 = mw[nb * NM_ + 1] * msk;
    const float s2 = mw[nb * NM_ + 2] * msk;
    const int orow = otile * 16 + n;            // A-fragment row (o index)
    const float2* Xp = Xm + (size_t)p * W_ * B_;

    v8f accR = {}; v8f accI = {};
    for (int k0 = 0; k0 < 16; ++k0) {
        const int i0 = k0 * 4 + kh * 2;
        const int i1 = i0 + 1;
        // X fragment (re,im) for this K pair, column b = n
        float2 x0 = Xp[i0 * B_ + nb];
        float2 x1 = Xp[i1 * B_ + nb];
        #pragma unroll
        for (int m = 0; m < NM_; ++m) {
            const float sm = (m == 0) ? s0 : ((m == 1) ? s1 : s2);
            v2f bx  = { x0.x * sm,  x1.x * sm };   // Re(X)*mw
            v2f by  = { x0.y * sm,  x1.y * sm };   // Im(X)*mw
            v2f nby = { -by[0], -by[1] };
            const size_t wbase = (((size_t)m * W_ + i0) * W_ + orow) * NMODES_ + p;
            const size_t istr  = (size_t)W_ * NMODES_;
            v2f ar = { wr[wbase], wr[wbase + istr] };
            v2f ai = { wi[wbase], wi[wbase + istr] };
            // Re: Wr*Xr - Wi*Xi ; Im: Wr*Xi + Wi*Xr
            accR = __builtin_amdgcn_wmma_f32_16x16x4_f32(false, ar, false, bx,  (short)0, accR, false, false);
            accR = __builtin_amdgcn_wmma_f32_16x16x4_f32(false, ai, false, nby, (short)0, accR, false, false);
            accI = __builtin_amdgcn_wmma_f32_16x16x4_f32(false, ar, false, by,  (short)0, accI, false, false);
            accI = __builtin_amdgcn_wmma_f32_16x16x4_f32(false, ai, false, bx,  (short)0, accI, false, false);
        }
    }
    // D layout: VGPR r -> M = r (lanes 0-15) / 8+r (lanes 16-31); N = lane&15
    if (n < 8) {
        #pragma unroll
        for (int r = 0; r < 8; ++r) {
            int oo = otile * 16 + kh * 8 + r;
            Ym[((size_t)n * W_ + oo) * NMODES_ + p] = make_float2(accR[r], accI[r]);
        }
    }
}

// ---------------------------------------------------------------------------
// K7: inverse partial DFT along x (16 freqs -> 32 samples). out D1[bo][x][ky][kz]
// ---------------------------------------------------------------------------
__global__ void k_idftx(const float2* __restrict__ Ym, float2* __restrict__ out) {
    __shared__ float2 tw[32];
    if (threadIdx.x < 32) {
        float a = TWOPI_OVER_32 * (float)threadIdx.x;
        tw[threadIdx.x] = make_float2(cosf(a), sinf(a));
    }
    __syncthreads();
    size_t id = (size_t)blockIdx.x * 256 + threadIdx.x;
    int kz = (int)(id & 15);
    int ky = (int)((id >> 4) & 15);
    int x  = (int)((id >> 8) & 31);
    int bo = (int)(id >> 13);
    const float2* base = Ym + (size_t)bo * NMODES_ + ky * MODE_ + kz;
    float ar = 0.f, ai = 0.f;
    #pragma unroll 8
    for (int kx = 0; kx < MODE_; ++kx) {
        float2 u = base[kx * MODE_ * MODE_];
        float2 w = tw[(kx * x) & 31];
        ar += u.x * w.x - u.y * w.y;
        ai += u.x * w.y + u.y * w.x;
    }
    out[id] = make_float2(ar, ai);
}

// ---------------------------------------------------------------------------
// K8: inverse partial DFT along y. out D2[bo][x][y][kz]
// ---------------------------------------------------------------------------
__global__ void k_idfty(const float2* __restrict__ in, float2* __restrict__ out) {
    __shared__ float2 tw[32];
    if (threadIdx.x < 32) {
        float a = TWOPI_OVER_32 * (float)threadIdx.x;
        tw[threadIdx.x] = make_float2(cosf(a), sinf(a));
    }
    __syncthreads();
    size_t id = (size_t)blockIdx.x * 256 + threadIdx.x;
    int kz = (int)(id & 15);
    int y  = (int)((id >> 4) & 31);
    int x  = (int)((id >> 9) & 31);
    int bo = (int)(id >> 14);
    const float2* base = in + (((size_t)bo * S_ + x) * MODE_) * MODE_ + kz;
    float ar = 0.f, ai = 0.f;
    #pragma unroll 8
    for (int ky = 0; ky < MODE_; ++ky) {
        float2 u = base[ky * MODE_];
        float2 w = tw[(ky * y) & 31];
        ar += u.x * w.x - u.y * w.y;
        ai += u.x * w.y + u.y * w.x;
    }
    out[id] = make_float2(ar, ai);
}

// ---------------------------------------------------------------------------
// K9: inverse real DFT along z (16 complex -> 32 real) + 1/N + GELU -> h
// ---------------------------------------------------------------------------
__global__ void k_idftz_gelu(const float2* __restrict__ in, float* __restrict__ h) {
    __shared__ float2 tw[32];
    if (threadIdx.x < 32) {
        float a = TWOPI_OVER_32 * (float)threadIdx.x;
        tw[threadIdx.x] = make_float2(cosf(a), sinf(a));
    }
    __syncthreads();
    size_t id = (size_t)blockIdx.x * 256 + threadIdx.x;
    int z  = (int)(id & 31);
    int y  = (int)((id >> 5) & 31);
    int x  = (int)((id >> 10) & 31);
    int bo = (int)(id >> 15);
    const float2* base = in + (((size_t)bo * S_ + x) * S_ + y) * MODE_;
    float acc = base[0].x;                       // kz = 0 term
    #pragma unroll 5
    for (int kz = 1; kz < MODE_; ++kz) {
        float2 u = base[kz];
        float2 w = tw[(kz * z) & 31];
        acc += 2.0f * (u.x * w.x - u.y * w.y);   // real-FFT hermitian fold
    }
    acc *= (1.0f / 32768.0f);                    // irfftn 1/N normalization
    h[id] = gelu_tanh(acc);
}

// ---------------------------------------------------------------------------
// host launcher
// ---------------------------------------------------------------------------
extern "C" void kernel_launch(void* const* d_in, const int* in_sizes, int n_in,
                              void* d_out, int out_size, void* d_ws, size_t ws_size,
                              hipStream_t stream) {
    const float* x      = (const float*)d_in[0];
    const float* emb_w  = (const float*)d_in[1];
    const float* emb_b  = (const float*)d_in[2];
    const float* gn_g   = (const float*)d_in[3];
    const float* gn_b   = (const float*)d_in[4];
    const float* att_w1 = (const float*)d_in[5];
    const float* att_b1 = (const float*)d_in[6];
    const float* att_w2 = (const float*)d_in[7];
    const float* att_b2 = (const float*)d_in[8];
    const float* wr     = (const float*)d_in[9];
    const float* wi     = (const float*)d_in[10];
    float* h = (float*)d_out;  // (B, W, S, S, S) fp32, persists across layers

    // workspace carve
    char* ws = (char*)d_ws;
    float2* C1 = (float2*)ws;                                   // 67,108,864 B
    float2* C2 = (float2*)(ws + 67108864);                      // 33,554,432 B
    float2* Xm = (float2*)(ws + 67108864 + 33554432);           // 16,777,216 B
    float2* Ym = (float2*)(ws + 67108864 + 33554432 + 16777216);// 16,777,216 B
    float*  st = (float*)(ws + 67108864 + 33554432 + 2 * 16777216);
    float* chanSum = st;               // 512
    float* chanSqs = st + 512;         // 512
    float* gnScale = st + 1024;        // 512
    float* gnShift = st + 1536;        // 512
    float* mwArr   = st + 2048;        // 24

    k_embed<<<dim3(S3_ / 256, B_), 256, 0, stream>>>(x, emb_w, emb_b, h);

    for (int l = 0; l < NL_; ++l) {
        const float* gam = gn_g   + (size_t)l * W_;
        const float* bet = gn_b   + (size_t)l * W_;
        const float* w1  = att_w1 + (size_t)l * 64 * W_;
        const float* b1  = att_b1 + (size_t)l * 64;
        const float* w2  = att_w2 + (size_t)l * NM_ * 64;
        const float* b2  = att_b2 + (size_t)l * NM_;
        const float* wrl = wr + (size_t)l * NM_ * W_ * W_ * NMODES_;
        const float* wil = wi + (size_t)l * NM_ * W_ * W_ * NMODES_;

        k_stats<<<B_ * W_, 256, 0, stream>>>(h, chanSum, chanSqs);
        k_gn_attn<<<1, 256, 0, stream>>>(chanSum, chanSqs, gam, bet, w1, b1, w2, b2,
                                         gnScale, gnShift, mwArr);
        k_dftz<<<(B_ * W_ * S_ * S_ * MODE_) / 256, 256, 0, stream>>>(h, gnScale, gnShift, C1);
        k_dfty<<<(B_ * W_ * S_ * MODE_ * MODE_) / 256, 256, 0, stream>>>(C1, C2);
        k_dftx<<<(B_ * W_ * NMODES_) / 256, 256, 0, stream>>>(C2, Xm);
        k_contract<<<dim3(NMODES_ / 32, 4), 1024, 0, stream>>>(Xm, wrl, wil, mwArr, Ym);
        k_idftx<<<(B_ * W_ * S_ * MODE_ * MODE_) / 256, 256, 0, stream>>>(Ym, C2);  // D1 = C2
        k_idfty<<<(B_ * W_ * S_ * S_ * MODE_) / 256, 256, 0, stream>>>(C2, C1);     // D2 = C1
        k_idftz_gelu<<<(B_ * W_ * S3_) / 256, 256, 0, stream>>>(C1, h);
    }
}